// LocSE_64965675319375
// MI455X (gfx1250) — compile-verified
//
#include <hip/hip_runtime.h>

typedef __attribute__((ext_vector_type(2))) float v2f;
typedef __attribute__((ext_vector_type(8))) float v8f;

#define NPTS 8192
#define KNN 16
#define NTILES (NPTS / 16)   // 512, power of two

// ---------------------------------------------------------------------------
// Kernel A: pack coords (B,N,3) -> float4 [x, y, z, |c|^2]
// ---------------------------------------------------------------------------
__global__ __launch_bounds__(256) void pack_kernel(const float* __restrict__ coords,
                                                   float4* __restrict__ pk) {
  int p = blockIdx.x * blockDim.x + threadIdx.x;   // 0..16383, exact fit
  float x = coords[p * 3 + 0];
  float y = coords[p * 3 + 1];
  float z = coords[p * 3 + 2];
  pk[p] = make_float4(x, y, z, x * x + y * y + z * z);
}

// ---------------------------------------------------------------------------
// Kernel B: brute-force KNN.
// Each wave owns 16 queries (queries along WMMA N / lanes).
// Per 16-candidate tile one V_WMMA_F32_16X16X4_F32 computes the 16x16 d2
// tile:  A row = [xc, yc, zc, |c|^2], B col = [-2xq, -2yq, -2zq, 1],
// C = |q|^2 broadcast per column.
// Lane keeps a private sorted top-16 in registers, updated by a BRANCHLESS
// min/max compare-exchange bubble (no exec-mask chains). Lanes L and L+16
// cover disjoint candidate rows; merged through LDS at the end.
// ---------------------------------------------------------------------------
__global__ __launch_bounds__(256) void knn_kernel(const float4* __restrict__ pk,
                                                  float* __restrict__ knn_d2,
                                                  int* __restrict__ knn_idx) {
  __shared__ float sd[256 * KNN];
  __shared__ int   si[256 * KNN];

  const int  lane  = threadIdx.x & 31;
  const int  wave  = threadIdx.x >> 5;
  const int  wgid  = blockIdx.x * 8 + wave;     // 0..1023 global wave id
  const int  qbase = wgid * 16;                 // first query of this wave
  const bool lo    = lane < 16;
  const int  q     = qbase + (lane & 15);       // this lane's query point
  const int  b     = q >> 13;                   // batch (8192 pts/batch)

  const float4 cq = pk[q];

  // B operand: col = lane&15; low lanes hold K=0,1; high lanes K=2,3
  v2f bmat;
  bmat.x = lo ? (-2.0f * cq.x) : (-2.0f * cq.z);
  bmat.y = lo ? (-2.0f * cq.y) : 1.0f;

  // C operand: |q|^2 for this lane's column, all 8 rows
  v8f cacc;
#pragma unroll
  for (int v = 0; v < 8; ++v) cacc[v] = cq.w;

  float dist[KNN];
  int   idx[KNN];
#pragma unroll
  for (int t = 0; t < KNN; ++t) { dist[t] = 3.4e38f; idx[t] = 0; }

  const int cb0 = b << 13;             // candidate base (same batch)
  const int hi8 = lo ? 0 : 8;          // D rows: low lanes 0..7, high 8..15
  const int lsel = lane & 15;

  // software pipeline: f holds tile t's candidate point
  float4 f = pk[cb0 + lsel];
  for (int t = 0; t < NTILES; ++t) {
    v2f a;                              // A row m = lane&15
    a.x = lo ? f.x : f.z;               // low: K=0,1 = x,y ; high: K=2,3 = z,sq
    a.y = lo ? f.y : f.w;

    // prefetch next tile (wraps on last iter; always in-bounds, no branch)
    f = pk[cb0 + ((t + 1) & (NTILES - 1)) * 16 + lsel];

    v8f d = __builtin_amdgcn_wmma_f32_16x16x4_f32(
        false, a, false, bmat, (short)0, cacc, false, false);

#pragma unroll
    for (int v = 0; v < 8; ++v) {
      float d2 = d[v];
      int   ci = t * 16 + v + hi8;      // candidate local index (within batch)
      if (d2 < dist[KNN - 1]) {
        // branchless bubble: carry (cd,cix) through the sorted list
        float cd  = d2;
        int   cix = ci;
#pragma unroll
        for (int s = 0; s < KNN; ++s) {
          bool  c    = cd < dist[s];
          float dmin = fminf(cd, dist[s]);
          float dmax = fmaxf(cd, dist[s]);
          int   imin = c ? cix : idx[s];
          int   imax = c ? idx[s] : cix;
          dist[s] = dmin;  idx[s] = imin;
          cd      = dmax;  cix    = imax;
        }
      }
    }
  }

  // dump per-lane lists to LDS, then merge lane L with lane L+16
  const int base = threadIdx.x * KNN;
#pragma unroll
  for (int t = 0; t < KNN; ++t) { sd[base + t] = dist[t]; si[base + t] = idx[t]; }
  __syncthreads();

  if (lo) {
    const int qa = threadIdx.x * KNN;
    const int qb = (threadIdx.x + 16) * KNN;
    int ia = 0, ib = 0;
    const long ob = (long)q * KNN;
    for (int k = 0; k < KNN; ++k) {     // ia<=k<16, ib<=k<16 -> never OOB
      float da = sd[qa + ia];
      float db = sd[qb + ib];
      bool  ta = da <= db;
      knn_d2[ob + k]  = ta ? da : db;
      knn_idx[ob + k] = ta ? si[qa + ia] : si[qb + ib];
      if (ta) ++ia; else ++ib;
    }
  }
}

// ---------------------------------------------------------------------------
// Kernel C: spatial encoding + 1x1 conv (W 16x10 via 3x WMMA f32 K=4, bias in C)
// + broadcast-feature concat. Each wave handles 16 (n,k) output columns.
// ---------------------------------------------------------------------------
__global__ __launch_bounds__(256) void encode_kernel(
    const float4* __restrict__ pk, const float* __restrict__ knn_d2,
    const int* __restrict__ knn_idx, const float* __restrict__ feat,
    const float* __restrict__ W, const float* __restrict__ bias,
    float* __restrict__ out) {
  const int  lane = threadIdx.x & 31;
  const bool lo   = lane < 16;
  const int  wave = (blockIdx.x * blockDim.x + threadIdx.x) >> 5; // 0..16383
  const int  col  = wave * 16 + (lane & 15);                      // 0..262143
  const int  k    = col & 15;
  const int  n    = (col >> 4) & (NPTS - 1);
  const int  b    = col >> 17;
  const int  q    = (b << 13) + n;

  float d2   = knn_d2[col];
  float dist = sqrtf(fmaxf(d2, 0.0f));
  int   ni   = knn_idx[col];

  float4 cc = pk[q];
  float4 cn = pk[(b << 13) + ni];

  // 10-channel encoding: [center, nbr, center-nbr, dist]
  float s[10] = {cc.x, cc.y, cc.z, cn.x, cn.y, cn.z,
                 cc.x - cn.x, cc.y - cn.y, cc.z - cn.z, dist};

  const int m  = lane & 15;     // A (W) row for this lane
  const int k0 = lo ? 0 : 2;    // low lanes K=4j..4j+1, high K=4j+2..4j+3

  v2f a0, a1, a2, bm0, bm1, bm2;
  a0.x = W[m * 10 + k0 + 0];  a0.y = W[m * 10 + k0 + 1];
  a1.x = W[m * 10 + k0 + 4];  a1.y = W[m * 10 + k0 + 5];
  a2.x = lo ? W[m * 10 + 8] : 0.0f;   // K=8,9 valid; K=10,11 padded with 0
  a2.y = lo ? W[m * 10 + 9] : 0.0f;

  bm0.x = s[k0 + 0];  bm0.y = s[k0 + 1];
  bm1.x = s[k0 + 4];  bm1.y = s[k0 + 5];
  bm2.x = lo ? s[8] : 0.0f;
  bm2.y = lo ? s[9] : 0.0f;

  const int rb = lo ? 0 : 8;    // D rows this lane holds
  v8f acc;
#pragma unroll
  for (int v = 0; v < 8; ++v) acc[v] = bias[rb + v];

  acc = __builtin_amdgcn_wmma_f32_16x16x4_f32(false, a0, false, bm0, (short)0, acc, false, false);
  acc = __builtin_amdgcn_wmma_f32_16x16x4_f32(false, a1, false, bm1, (short)0, acc, false, false);
  acc = __builtin_amdgcn_wmma_f32_16x16x4_f32(false, a2, false, bm2, (short)0, acc, false, false);

  // channels 0..15: mlp_enc
#pragma unroll
  for (int v = 0; v < 8; ++v) {
    int  ch = rb + v;
    long o  = ((long)(b * 32 + ch) * NPTS + n) * 16 + k;
    out[o]  = acc[v];
  }
  // channels 16..31: broadcast features
#pragma unroll
  for (int v = 0; v < 8; ++v) {
    int   cf = rb + v;
    float fv = feat[(b * 16 + cf) * NPTS + n];
    long  o  = ((long)(b * 32 + 16 + cf) * NPTS + n) * 16 + k;
    out[o]   = fv;
  }
}

// ---------------------------------------------------------------------------
extern "C" void kernel_launch(void* const* d_in, const int* in_sizes, int n_in,
                              void* d_out, int out_size, void* d_ws, size_t ws_size,
                              hipStream_t stream) {
  const float* coords = (const float*)d_in[0];  // (2, 8192, 3)
  const float* feat   = (const float*)d_in[1];  // (2, 16, 8192, 1)
  const float* W      = (const float*)d_in[2];  // (16, 10)
  const float* bias   = (const float*)d_in[3];  // (16,)
  float* out = (float*)d_out;                   // (2, 32, 8192, 16)

  char*   ws      = (char*)d_ws;
  float4* pk      = (float4*)ws;                           // 16384*16 B = 256 KB
  float*  knn_d2  = (float*)(ws + 262144);                 // 262144*4 = 1 MB
  int*    knn_idx = (int*)(ws + 262144 + 1048576);         // 1 MB

  pack_kernel<<<64, 256, 0, stream>>>(coords, pk);
  knn_kernel<<<128, 256, 0, stream>>>(pk, knn_d2, knn_idx);
  encode_kernel<<<2048, 256, 0, stream>>>(pk, knn_d2, knn_idx, feat, W, bias, out);
}